// NpairLoss_12807592476698
// MI455X (gfx1250) — compile-verified
//
#include <hip/hip_runtime.h>
#include <math.h>
#include <stdint.h>

typedef __attribute__((ext_vector_type(2))) float    v2f;
typedef __attribute__((ext_vector_type(8))) float    v8f;
typedef __attribute__((ext_vector_type(4))) unsigned u32x4;
typedef __attribute__((ext_vector_type(8))) unsigned u32x8;

#define NROWS 8192
#define CDIM  64
#define SCALE2 0.125f        // 64^-0.5
#define LDS_ROW 66           // 64 DW row + 2 DW TDM pad (keeps b64 8B-aligned)
#define TILE_DW (16 * LDS_ROW)

// ---------------------------------------------------------------------------
// K0: zero the sumexp accumulator
// ---------------------------------------------------------------------------
__global__ void __launch_bounds__(256) k_zero(float* __restrict__ p) {
    p[blockIdx.x * blockDim.x + threadIdx.x] = 0.f;
}

// ---------------------------------------------------------------------------
// K1: nrm2[i] = scale^2 * sum_c I[i][c]^2   (== diagonal of scaled Gram)
// ---------------------------------------------------------------------------
__global__ void __launch_bounds__(256) k_rownorm2(const float* __restrict__ I,
                                                  float* __restrict__ nrm2) {
    int row = blockIdx.x * blockDim.x + threadIdx.x;
    if (row >= NROWS) return;
    const float4* p = (const float4*)(I + (size_t)row * CDIM);
    float s = 0.f;
#pragma unroll
    for (int i = 0; i < CDIM / 4; ++i) {
        float4 v = p[i];
        s += v.x * v.x + v.y * v.y + v.z * v.z + v.w * v.w;
    }
    nrm2[row] = s * SCALE2;
}

// ---------------------------------------------------------------------------
// TDM: DMA one 16x64 fp32 j-tile (contiguous 4096 floats) into LDS,
// inserting a 2-DWORD pad after every 64 DWORDs (row stride 66 DW in LDS).
// D# built per CDNA5 ISA; 2-group form (<=2D tensor).
// ---------------------------------------------------------------------------
__device__ __forceinline__ void tdm_load_tile(uint32_t lds_byte, const float* gptr) {
    uint64_t ga = (uint64_t)(uintptr_t)gptr;
    u32x4 g0;
    g0.x = 1u;                                   // count=1 valid descriptor
    g0.y = lds_byte;                             // lds_addr
    g0.z = (uint32_t)ga;                         // global_addr[31:0]
    g0.w = (uint32_t)((ga >> 32) & 0x01FFFFFFu)  // global_addr[56:32]
           | (2u << 30);                         // type=2 ("image")
    u32x8 g1;
    g1[0] = (2u << 16)      // data_size = 4 bytes
          | (1u << 20)      // pad_enable
          | (5u << 22)      // pad_interval: 64 DWORDs
          | (1u << 25);     // pad_amount: 2 DWORDs
    g1[1] = (4096u << 16);  // tensor_dim0[15:0] = 4096 elements
    g1[2] = (1u << 16);     // tensor_dim0[31:16]=0 | tensor_dim1=1
    g1[3] = (4096u << 16);  // tile_dim0 = 4096 elements (1-D tile)
    g1[4] = 0u;             // tile_dim1=0 (unused), tile_dim2=0
    g1[5] = 4096u;          // tensor_dim0_stride[31:0]
    g1[6] = 0u;
    g1[7] = 0u;
    asm volatile("tensor_load_to_lds %0, %1" : : "s"(g0), "s"(g1) : "memory");
}

// ---------------------------------------------------------------------------
// K2: WMMA Gram tiles + row-wise sum(exp(dot - diag)).
// Grid (64, 8), block = 256 threads (8 waves).
//   blockIdx.x -> 128-row i-slab (wave w owns i-tile rows x*128+w*16 ..+16)
//   blockIdx.y -> 1024-col j-slab (64 j-tiles), all 8 waves share each tile.
// j-tiles staged into LDS by the Tensor Data Mover, double-buffered.
// Partial row sums combined with global_atomic_add_f32.
// ---------------------------------------------------------------------------
__global__ void __launch_bounds__(256) k_sumexp(const float* __restrict__ I,
                                                const float* __restrict__ nrm2,
                                                float* __restrict__ sumexp) {
    __shared__ float sbuf[2][TILE_DW];

    const int tid  = threadIdx.x;
    const int wave = tid >> 5;
    const int lane = tid & 31;
    const int half = lane >> 4;   // K sub-pair select
    const int mr   = lane & 15;   // row within A tile / col within B tile
    const int iBase  = blockIdx.x * 128 + wave * 16;
    const int jTile0 = blockIdx.y * 64;

    const uint32_t ldsBase = (uint32_t)(uintptr_t)(&sbuf[0][0]);

    // --- A fragment: 16 K-steps of v2f, scale^2 folded in (loaded once) ---
    const float* arow = I + (size_t)(iBase + mr) * CDIM + 2 * half;
    v2f a[16];
#pragma unroll
    for (int kk = 0; kk < 16; ++kk) {
        v2f t = *(const v2f*)(arow + kk * 4);
        a[kk] = t * SCALE2;
    }

    // Diagonal values: C reg v <-> row (v + 8*half) of this wave's tile.
    float diag[8];
#pragma unroll
    for (int v = 0; v < 8; ++v) diag[v] = nrm2[iBase + v + 8 * half];

    float sums[8] = {0.f, 0.f, 0.f, 0.f, 0.f, 0.f, 0.f, 0.f};

    // Prologue: stage first tile.
    if (wave == 0) {
        tdm_load_tile(ldsBase, I + (size_t)(jTile0 * 16) * CDIM);
        __builtin_amdgcn_s_wait_tensorcnt(0);
    }
    __syncthreads();

    for (int jt = 0; jt < 64; ++jt) {
        const int cur = jt & 1;

        // Prefetch next tile into the other buffer while we compute.
        if (wave == 0 && jt + 1 < 64) {
            tdm_load_tile(ldsBase + (uint32_t)((cur ^ 1) * TILE_DW * 4),
                          I + (size_t)((jTile0 + jt + 1) * 16) * CDIM);
        }

        // B fragments from LDS (padded row stride keeps b64 aligned).
        const float* bbase = &sbuf[cur][0] + mr * LDS_ROW + 2 * half;
        v2f b[16];
#pragma unroll
        for (int kk = 0; kk < 16; ++kk)
            b[kk] = *(const v2f*)(bbase + kk * 4);

        v8f acc0 = {};
        v8f acc1 = {};
#pragma unroll
        for (int kk = 0; kk < 16; kk += 2) {
            acc0 = __builtin_amdgcn_wmma_f32_16x16x4_f32(
                false, a[kk],     false, b[kk],     (short)0, acc0, false, false);
            acc1 = __builtin_amdgcn_wmma_f32_16x16x4_f32(
                false, a[kk + 1], false, b[kk + 1], (short)0, acc1, false, false);
        }

#pragma unroll
        for (int v = 0; v < 8; ++v)
            sums[v] += __expf(acc0[v] + acc1[v] - diag[v]);

        // Drain TDM before anyone touches the freshly staged buffer.
        if (wave == 0) __builtin_amdgcn_s_wait_tensorcnt(0);
        __syncthreads();
    }

    // Reduce each C register across its 16-lane group, then accumulate
    // globally (blockIdx.y slabs share rows).
#pragma unroll
    for (int v = 0; v < 8; ++v) {
        float s = sums[v];
        s += __shfl_xor(s, 1, 32);
        s += __shfl_xor(s, 2, 32);
        s += __shfl_xor(s, 4, 32);
        s += __shfl_xor(s, 8, 32);
        if (mr == 0) atomicAdd(&sumexp[iBase + v + 8 * half], s);
    }
}

// ---------------------------------------------------------------------------
// K3: rowlog[i] = sum_j log(ir[i][j]) with
//     ir[i][j] = (i==j) ? 0 : -(n_i*n_j + n_i^2),  n = sqrt(nrm2)
// ---------------------------------------------------------------------------
__global__ void __launch_bounds__(256) k_rowlog(const float* __restrict__ nrm2,
                                                float* __restrict__ rowlog) {
    __shared__ float snrm[NROWS];
    for (int i = threadIdx.x; i < NROWS; i += blockDim.x)
        snrm[i] = sqrtf(nrm2[i]);
    __syncthreads();

    int row = blockIdx.x * blockDim.x + threadIdx.x;
    if (row >= NROWS) return;
    float ni  = snrm[row];
    float ni2 = ni * ni;
    float s = 0.f;
    for (int j = 0; j < NROWS; ++j) {
        float v = (j == row) ? 0.0f : (-(ni * snrm[j] + ni2));
        s += __logf(v);
    }
    rowlog[row] = s;
}

// ---------------------------------------------------------------------------
// K4: loss = sum(log(sumexp))/N^2 - sum(rowlog)/N^3
// ---------------------------------------------------------------------------
__global__ void __launch_bounds__(256) k_final(const float* __restrict__ sumexp,
                                               const float* __restrict__ rowlog,
                                               float* __restrict__ out) {
    __shared__ float red[256];
    float sl = 0.f, sr = 0.f;
    for (int i = threadIdx.x; i < NROWS; i += 256) {
        sl += __logf(sumexp[i]);
        sr += rowlog[i];
    }
    red[threadIdx.x] = sl;
    __syncthreads();
    for (int s = 128; s > 0; s >>= 1) {
        if (threadIdx.x < s) red[threadIdx.x] += red[threadIdx.x + s];
        __syncthreads();
    }
    float totL = red[0];
    __syncthreads();
    red[threadIdx.x] = sr;
    __syncthreads();
    for (int s = 128; s > 0; s >>= 1) {
        if (threadIdx.x < s) red[threadIdx.x] += red[threadIdx.x + s];
        __syncthreads();
    }
    if (threadIdx.x == 0) {
        double Nf = (double)NROWS;
        double loss = (double)totL / (Nf * Nf) - (double)red[0] / (Nf * Nf * Nf);
        out[0] = (float)loss;
    }
}

// ---------------------------------------------------------------------------
extern "C" void kernel_launch(void* const* d_in, const int* in_sizes, int n_in,
                              void* d_out, int out_size, void* d_ws, size_t ws_size,
                              hipStream_t stream) {
    (void)in_sizes; (void)n_in; (void)out_size; (void)ws_size;
    const float* I = (const float*)d_in[0];
    float* out = (float*)d_out;

    float* nrm2   = (float*)d_ws;        // 8192 floats
    float* sumexp = nrm2 + NROWS;        // 8192 floats
    float* rowlog = sumexp + NROWS;      // 8192 floats

    k_zero<<<NROWS / 256, 256, 0, stream>>>(sumexp);
    k_rownorm2<<<NROWS / 256, 256, 0, stream>>>(I, nrm2);
    k_sumexp<<<dim3(NROWS / 128, 8), 256, 0, stream>>>(I, nrm2, sumexp);
    k_rowlog<<<NROWS / 256, 256, 0, stream>>>(nrm2, rowlog);
    k_final<<<1, 256, 0, stream>>>(sumexp, rowlog, out);
}